// Wave_Order_Recognition_SNN_85899345920125
// MI455X (gfx1250) — compile-verified
//
#include <hip/hip_runtime.h>
#include <hip/hip_bf16.h>

// Problem dims (fixed by the reference)
#define B_TOT 2048
#define T_TOT 120
#define H_DIM 512
#define O_DIM 2
#define BT 16          // batch tile per workgroup
#define NTHREADS 512   // 16 waves of 32
#define NWAVES 16
#define SPK_PITCH 516  // padded LDS row pitch (floats) to spread banks

typedef float v2f __attribute__((ext_vector_type(2)));
typedef float v8f __attribute__((ext_vector_type(8)));

__global__ void snn_init_ws(unsigned* ws) {
  if (threadIdx.x < 2) ws[threadIdx.x] = 0u;
}

__global__ __launch_bounds__(NTHREADS)
void snn_main(const float* __restrict__ x,       // (B,T)
              const float* __restrict__ W1,      // (H,1)
              const float* __restrict__ W2,      // (O,H)
              const float* __restrict__ beta_h,  // (H)
              const float* __restrict__ thr_h,   // (H)
              const float* __restrict__ beta_o,  // (O)
              const float* __restrict__ thr_o,   // (O)
              float* __restrict__ out_spk1,      // (T,B,H)
              float* __restrict__ out_mem1,      // (T,B,H)
              float* __restrict__ out_spk2,      // (T,B,O)
              float* __restrict__ out_mem2,      // (T,B,O)
              unsigned* __restrict__ ws_counts)
{
  __shared__ float lds_x[BT * T_TOT];          // x tile, broadcast reads
  __shared__ float lds_spk[BT * SPK_PITCH];    // spike tile (16 x 512, padded)
  __shared__ float lds_part[NWAVES * 32];      // per-wave partial cur2 (16b x 2o)
  __shared__ unsigned s_hcnt, s_ocnt;

  const int tid = threadIdx.x;                 // hidden neuron h = tid
  const int b0  = blockIdx.x * BT;

  if (tid == 0) { s_hcnt = 0u; s_ocnt = 0u; }

  // Preload x tile into LDS
  for (int idx = tid; idx < BT * T_TOT; idx += NTHREADS) {
    int i = idx / T_TOT, tt = idx % T_TOT;
    lds_x[i * T_TOT + tt] = x[(size_t)(b0 + i) * T_TOT + tt];
  }

  // Per-thread hidden-neuron constants
  const float w1h = W1[tid];
  float bh = beta_h[tid]; bh = fminf(fmaxf(bh, 0.f), 1.f);
  const float thh = thr_h[tid];

  // Wave identity; preload this wave's W2 B-fragments (constant over time).
  // B (4x16 f32) layout: VGPR0 lanes0-15 = row K+0, lanes16-31 = row K+2;
  // VGPR1 = rows K+1 / K+3. Column n = output o (cols >= O are zero-padded).
  const int w = tid >> 5;
  const int l = tid & 31;
  const int n = l & 15;
  const int khalf = (l >= 16) ? 2 : 0;
  v2f bfr[8];
  #pragma unroll
  for (int k = 0; k < 8; ++k) {
    int K = w * 32 + k * 4 + khalf;
    float e0 = 0.f, e1 = 0.f;
    if (n < O_DIM) { e0 = W2[n * H_DIM + K]; e1 = W2[n * H_DIM + K + 1]; }
    bfr[k] = (v2f){e0, e1};
  }

  // Wave-0 per-lane output-neuron constants: lane l -> (b = l>>1, o = l&1)
  const int oo = l & 1;
  float bo = beta_o[oo]; bo = fminf(fmaxf(bo, 0.f), 1.f);
  const float tho = thr_o[oo];

  float mem1[BT];
  #pragma unroll
  for (int i = 0; i < BT; ++i) mem1[i] = 0.f;
  float m2 = 0.f;
  unsigned hcnt = 0u, ocnt = 0u;

  __syncthreads();

  for (int t = 0; t < T_TOT; ++t) {
    // ---- Phase 1: hidden LIF update + coalesced record stores ----
    const size_t base1 = (size_t)t * ((size_t)B_TOT * H_DIM)
                       + (size_t)b0 * H_DIM + (size_t)tid;
    #pragma unroll
    for (int i = 0; i < BT; ++i) {
      float cur = lds_x[i * T_TOT + t] * w1h;     // LDS broadcast
      float m = bh * mem1[i] + cur;
      bool fire = (m > thh);
      float spk = fire ? 1.f : 0.f;
      m = fire ? 0.f : m;                          // reset-to-zero
      out_spk1[base1 + (size_t)i * H_DIM] = spk;
      out_mem1[base1 + (size_t)i * H_DIM] = m;
      lds_spk[i * SPK_PITCH + tid] = spk;
      hcnt += fire ? 1u : 0u;
      mem1[i] = m;
    }
    __syncthreads();

    // ---- Phase 2: every wave does its K-slice of cur2 = spk1 @ W2^T ----
    // A (16x4 f32) layout: lanes0-15 row M=l, K+0/K+1 in the 2 VGPRs;
    // lanes16-31 row M=l-16, K+2/K+3.  (EXEC all-ones: no divergence here.)
    {
      const float* arow = &lds_spk[(l & 15) * SPK_PITCH];
      v8f acc = (v8f){0.f,0.f,0.f,0.f,0.f,0.f,0.f,0.f};
      #pragma unroll
      for (int k = 0; k < 8; ++k) {
        int kb = w * 32 + k * 4 + khalf;           // even -> 8B aligned
        v2f a = *(const v2f*)(arow + kb);
        acc = __builtin_amdgcn_wmma_f32_16x16x4_f32(
            false, a, false, bfr[k], (short)0, acc, false, false);
      }
      // Extract D columns 0..O-1.  D layout: VGPR j holds rows M=j (lanes0-15)
      // and M=j+8 (lanes16-31), column N = lane (mod 16).
      if (l < O_DIM) {
        #pragma unroll
        for (int j = 0; j < 8; ++j)
          lds_part[w * 32 + j * 2 + l] = acc[j];
      } else if (l >= 16 && l < 16 + O_DIM) {
        #pragma unroll
        for (int j = 0; j < 8; ++j)
          lds_part[w * 32 + (j + 8) * 2 + (l - 16)] = acc[j];
      }
    }
    __syncthreads();

    // ---- Phase 3: wave 0 reduces 16 partials, runs output LIF, records ----
    if (tid < 32) {
      float cur2 = 0.f;
      #pragma unroll
      for (int ww = 0; ww < NWAVES; ++ww) cur2 += lds_part[ww * 32 + tid];
      m2 = bo * m2 + cur2;
      bool fire2 = (m2 > tho);
      float spk2 = fire2 ? 1.f : 0.f;
      m2 = fire2 ? 0.f : m2;
      size_t base2 = (size_t)t * ((size_t)B_TOT * O_DIM)
                   + (size_t)b0 * O_DIM + (size_t)tid;  // 32 contiguous floats
      out_spk2[base2] = spk2;
      out_mem2[base2] = m2;
      ocnt += fire2 ? 1u : 0u;
    }
    __syncthreads();
  }

  // Exact integer spike counts (deterministic across replays)
  atomicAdd(&s_hcnt, hcnt);
  if (tid < 32) atomicAdd(&s_ocnt, ocnt);
  __syncthreads();
  if (tid == 0) {
    atomicAdd(&ws_counts[0], s_hcnt);
    atomicAdd(&ws_counts[1], s_ocnt);
  }
}

__global__ void snn_finalize(const unsigned* __restrict__ ws,
                             float* __restrict__ out_counts) {
  if (threadIdx.x == 0) {
    out_counts[0] = (float)ws[0];
    out_counts[1] = (float)ws[1];
  }
}

extern "C" void kernel_launch(void* const* d_in, const int* in_sizes, int n_in,
                              void* d_out, int out_size, void* d_ws, size_t ws_size,
                              hipStream_t stream) {
  const float* x      = (const float*)d_in[0];
  const float* W1     = (const float*)d_in[1];
  const float* W2     = (const float*)d_in[2];
  const float* beta_h = (const float*)d_in[3];
  const float* thr_h  = (const float*)d_in[4];
  const float* beta_o = (const float*)d_in[5];
  const float* thr_o  = (const float*)d_in[6];

  float* out = (float*)d_out;
  const size_t sz1 = (size_t)T_TOT * B_TOT * H_DIM;   // 125,829,120
  const size_t sz2 = (size_t)T_TOT * B_TOT * O_DIM;   // 491,520
  float* out_spk1   = out;
  float* out_mem1   = out + sz1;
  float* out_spk2   = out + 2 * sz1;
  float* out_mem2   = out + 2 * sz1 + sz2;
  float* out_counts = out + 2 * sz1 + 2 * sz2;

  unsigned* ws = (unsigned*)d_ws;

  snn_init_ws<<<1, 32, 0, stream>>>(ws);
  snn_main<<<B_TOT / BT, NTHREADS, 0, stream>>>(
      x, W1, W2, beta_h, thr_h, beta_o, thr_o,
      out_spk1, out_mem1, out_spk2, out_mem2, ws);
  snn_finalize<<<1, 32, 0, stream>>>(ws, out_counts);
}